// AdaptiveMask_66675072303610
// MI455X (gfx1250) — compile-verified
//
#include <hip/hip_runtime.h>
#include <hip/hip_bf16.h>

// ---------------------------------------------------------------------------
// MI455X (gfx1250) implementation.
// Bandwidth-bound streaming problem (~1.7 GB moved, ~75us floor @ 23.3 TB/s).
// Hot kernel uses CDNA5 async global->LDS copies (GLOBAL_LOAD_ASYNC_TO_LDS_B128,
// ASYNCcnt) for a per-wave double-buffered pipeline, bank-conflict-free
// diagonal ds_load_b128 reads of the staged tiles, and NT temporal hints so
// the single-use embedding stream doesn't thrash L2 (which holds the reused
// row_sum table / index lists / alpha array instead).
// ---------------------------------------------------------------------------

#define N_NODES 100000
#define TPW 8   // tiles (of 32 edges) per wave
#define WPB 2   // waves per block (64 threads, 32 KB static LDS)

// Copy 8 KB (32 rows x 256 B) from global (sbase + lane*16 + imm) to
// LDS (ldsaddr + imm). Same immediate is added to both sides by HW.
// 16 x b128 per wave-instruction group = 512 B per instruction, fully dense.
// NT hint: stream-once data, bypass temporal residency in L2.
#define ASYNC_COPY_8K(ldsaddr, voff, sbase)                                   \
  asm volatile(                                                               \
      "global_load_async_to_lds_b128 %0, %1, %2 offset:0 th:TH_LOAD_NT\n\t"   \
      "global_load_async_to_lds_b128 %0, %1, %2 offset:512 th:TH_LOAD_NT\n\t" \
      "global_load_async_to_lds_b128 %0, %1, %2 offset:1024 th:TH_LOAD_NT\n\t"\
      "global_load_async_to_lds_b128 %0, %1, %2 offset:1536 th:TH_LOAD_NT\n\t"\
      "global_load_async_to_lds_b128 %0, %1, %2 offset:2048 th:TH_LOAD_NT\n\t"\
      "global_load_async_to_lds_b128 %0, %1, %2 offset:2560 th:TH_LOAD_NT\n\t"\
      "global_load_async_to_lds_b128 %0, %1, %2 offset:3072 th:TH_LOAD_NT\n\t"\
      "global_load_async_to_lds_b128 %0, %1, %2 offset:3584 th:TH_LOAD_NT\n\t"\
      "global_load_async_to_lds_b128 %0, %1, %2 offset:4096 th:TH_LOAD_NT\n\t"\
      "global_load_async_to_lds_b128 %0, %1, %2 offset:4608 th:TH_LOAD_NT\n\t"\
      "global_load_async_to_lds_b128 %0, %1, %2 offset:5120 th:TH_LOAD_NT\n\t"\
      "global_load_async_to_lds_b128 %0, %1, %2 offset:5632 th:TH_LOAD_NT\n\t"\
      "global_load_async_to_lds_b128 %0, %1, %2 offset:6144 th:TH_LOAD_NT\n\t"\
      "global_load_async_to_lds_b128 %0, %1, %2 offset:6656 th:TH_LOAD_NT\n\t"\
      "global_load_async_to_lds_b128 %0, %1, %2 offset:7168 th:TH_LOAD_NT\n\t"\
      "global_load_async_to_lds_b128 %0, %1, %2 offset:7680 th:TH_LOAD_NT"    \
      :: "v"(ldsaddr), "v"(voff), "s"(sbase) : "memory")

__global__ void zero_kernel(float* __restrict__ p, int n) {
  int i = blockIdx.x * blockDim.x + threadIdx.x;
  if (i < n) p[i] = 0.0f;
}

__global__ __launch_bounds__(WPB * 32, 1)
void edge_alpha_kernel(const float* __restrict__ hE, const float* __restrict__ tE,
                       const int* __restrict__ head,
                       float* __restrict__ alphaOut, float* __restrict__ rowSum,
                       int numTiles) {
  // [warp][slot][h/t][32 rows * 64 floats]
  __shared__ float sh[WPB][2][2][32 * 64];

  const int lane = threadIdx.x & 31;
  const int warp = threadIdx.x >> 5;
  const int wid  = blockIdx.x * WPB + warp;

  int t0 = wid * TPW;
  if (t0 >= numTiles) return;
  int n = numTiles - t0;
  if (n > TPW) n = TPW;

  const unsigned voff = (unsigned)lane * 16u;
  unsigned ldsH[2], ldsT[2];
  ldsH[0] = (unsigned)(size_t)(&sh[warp][0][0][0]) + voff;
  ldsT[0] = (unsigned)(size_t)(&sh[warp][0][1][0]) + voff;
  ldsH[1] = (unsigned)(size_t)(&sh[warp][1][0][0]) + voff;
  ldsT[1] = (unsigned)(size_t)(&sh[warp][1][1][0]) + voff;

  const unsigned long long hBase = (unsigned long long)(size_t)hE;
  const unsigned long long tBase = (unsigned long long)(size_t)tE;

  auto issue = [&](int tile, int slot) {
    unsigned long long off = (unsigned long long)tile * 8192ull;
    ASYNC_COPY_8K(ldsH[slot], voff, hBase + off);
    ASYNC_COPY_8K(ldsT[slot], voff, tBase + off);
  };

  issue(t0, 0);
  if (n > 1) issue(t0 + 1, 1);

  for (int k = 0; k < n; ++k) {
    if (k + 1 < n) {
      asm volatile("s_wait_asynccnt 32" ::: "memory");  // tile k landed
    } else {
      asm volatile("s_wait_asynccnt 0" ::: "memory");   // drain
    }
    const int slot = k & 1;
    const float* hr = &sh[warp][slot][0][lane * 64];
    const float* tr = &sh[warp][slot][1][lane * 64];

    float hh = 0.0f, ht = 0.0f, tt = 0.0f;
    // Diagonal chunk order: lane i reads 16B chunk (i+t)&15 -> all 16 lanes of
    // each half-wave hit distinct 4-bank clusters (conflict-free ds_load_b128).
    #pragma unroll
    for (int t = 0; t < 16; ++t) {
      const int c = (lane + t) & 15;
      const float4 hv = *(const float4*)(hr + c * 4);
      const float4 tv = *(const float4*)(tr + c * 4);
      hh = fmaf(hv.x, hv.x, hh); hh = fmaf(hv.y, hv.y, hh);
      hh = fmaf(hv.z, hv.z, hh); hh = fmaf(hv.w, hv.w, hh);
      tt = fmaf(tv.x, tv.x, tt); tt = fmaf(tv.y, tv.y, tt);
      tt = fmaf(tv.z, tv.z, tt); tt = fmaf(tv.w, tv.w, tt);
      ht = fmaf(hv.x, tv.x, ht); ht = fmaf(hv.y, tv.y, ht);
      ht = fmaf(hv.z, tv.z, ht); ht = fmaf(hv.w, tv.w, ht);
    }
    const float nh = fmaxf(sqrtf(hh), 1e-12f);
    const float nt = fmaxf(sqrtf(tt), 1e-12f);
    const float a  = 0.5f * (ht / (nh * nt) + 1.0f);

    const int e = (t0 + k) * 32 + lane;
    alphaOut[e] = a;
    atomicAdd(&rowSum[head[e]], a);  // global_atomic_add_f32, no return

    if (k + 2 < n) issue(t0 + k + 2, slot);  // refill freed slot
  }
}

// Scalar fallback for E % 32 leftover edges (none for E = 3.2M).
__global__ void edge_alpha_tail(const float* __restrict__ hE, const float* __restrict__ tE,
                                const int* __restrict__ head,
                                float* __restrict__ alphaOut, float* __restrict__ rowSum,
                                int start, int E) {
  int e = start + blockIdx.x * blockDim.x + threadIdx.x;
  if (e >= E) return;
  const float4* h4 = (const float4*)(hE + (size_t)e * 64);
  const float4* t4 = (const float4*)(tE + (size_t)e * 64);
  float hh = 0.0f, ht = 0.0f, tt = 0.0f;
  #pragma unroll
  for (int i = 0; i < 16; ++i) {
    float4 hv = h4[i], tv = t4[i];
    hh = fmaf(hv.x, hv.x, hh); hh = fmaf(hv.y, hv.y, hh);
    hh = fmaf(hv.z, hv.z, hh); hh = fmaf(hv.w, hv.w, hh);
    tt = fmaf(tv.x, tv.x, tt); tt = fmaf(tv.y, tv.y, tt);
    tt = fmaf(tv.z, tv.z, tt); tt = fmaf(tv.w, tv.w, tt);
    ht = fmaf(hv.x, tv.x, ht); ht = fmaf(hv.y, tv.y, ht);
    ht = fmaf(hv.z, tv.z, ht); ht = fmaf(hv.w, tv.w, ht);
  }
  const float nh = fmaxf(sqrtf(hh), 1e-12f);
  const float nt = fmaxf(sqrtf(tt), 1e-12f);
  const float a  = 0.5f * (ht / (nh * nt) + 1.0f);
  alphaOut[e] = a;
  atomicAdd(&rowSum[head[e]], a);
}

// G_indices (2E int32) + G_values (E f32), x4 vectorized.
__global__ void finalize_kernel(const int* __restrict__ head, const int* __restrict__ tail,
                                const float* __restrict__ alpha, const float* __restrict__ rowSum,
                                int* __restrict__ outIdx, float* __restrict__ outVal, int E) {
  const int e = (blockIdx.x * blockDim.x + threadIdx.x) * 4;
  if (e + 3 < E) {
    const int4   h = *(const int4*)(head + e);
    const int4   t = *(const int4*)(tail + e);
    const float4 a = *(const float4*)(alpha + e);
    float4 v;
    float r;
    r = rowSum[h.x]; v.x = a.x * ((r != 0.0f) ? (1.0f / r) : 0.0f);
    r = rowSum[h.y]; v.y = a.y * ((r != 0.0f) ? (1.0f / r) : 0.0f);
    r = rowSum[h.z]; v.z = a.z * ((r != 0.0f) ? (1.0f / r) : 0.0f);
    r = rowSum[h.w]; v.w = a.w * ((r != 0.0f) ? (1.0f / r) : 0.0f);
    *(int4*)(outIdx + e)             = h;
    *(int4*)(outIdx + (size_t)E + e) = t;
    *(float4*)(outVal + e)           = v;
  } else {
    for (int i = e; i < E; ++i) {
      const int h = head[i];
      const float r = rowSum[h];
      outIdx[i]             = h;
      outIdx[(size_t)E + i] = tail[i];
      outVal[i] = alpha[i] * ((r != 0.0f) ? (1.0f / r) : 0.0f);
    }
  }
}

extern "C" void kernel_launch(void* const* d_in, const int* in_sizes, int n_in,
                              void* d_out, int out_size, void* d_ws, size_t ws_size,
                              hipStream_t stream) {
  const float* hE   = (const float*)d_in[0];
  const float* tE   = (const float*)d_in[1];
  const int*   head = (const int*)d_in[2];
  const int*   tail = (const int*)d_in[3];
  const int E = in_sizes[2];

  float* rowSum = (float*)d_ws;            // [N_NODES]
  float* alpha  = rowSum + N_NODES;        // [E]

  zero_kernel<<<(N_NODES + 255) / 256, 256, 0, stream>>>(rowSum, N_NODES);

  const int numTiles = E / 32;
  if (numTiles > 0) {
    const int waves  = (numTiles + TPW - 1) / TPW;
    const int blocks = (waves + WPB - 1) / WPB;
    edge_alpha_kernel<<<blocks, WPB * 32, 0, stream>>>(hE, tE, head, alpha, rowSum, numTiles);
  }
  const int rem = E - numTiles * 32;
  if (rem > 0) {
    edge_alpha_tail<<<(rem + 63) / 64, 64, 0, stream>>>(hE, tE, head, alpha, rowSum,
                                                        numTiles * 32, E);
  }

  int*   outIdx = (int*)d_out;
  float* outVal = (float*)d_out + 2 * (size_t)E;
  const int q = (E + 3) / 4;
  finalize_kernel<<<(q + 255) / 256, 256, 0, stream>>>(head, tail, alpha, rowSum,
                                                       outIdx, outVal, E);
}